// ModelNew_25056839205021
// MI455X (gfx1250) — compile-verified
//
#include <hip/hip_runtime.h>

typedef float v2f __attribute__((ext_vector_type(2)));
typedef float v8f __attribute__((ext_vector_type(8)));

#define N_TOK   49
#define D_HEAD  64
#define H_NUM   16
#define LSTR    68          // LDS row stride (floats): bank = (4r + k) mod 64, conflict-free
#define SCALE_Q 0.125f      // 1/sqrt(64)

__global__ __launch_bounds__(512) void swin_attn_f32wmma(
    const float* __restrict__ Q, const float* __restrict__ K,
    const float* __restrict__ V, const float* __restrict__ rpb,
    const int* __restrict__ rpi, float* __restrict__ out)
{
    __shared__ float smem[3 * 64 * LSTR];
    float* QVs = smem;               // Q (phase 1) then V^T (phase 2/3)
    float* Ks  = smem + 64 * LSTR;
    float* Ps  = smem + 2 * 64 * LSTR;

    const int    tid  = threadIdx.x;
    const int    bh   = blockIdx.x;
    const int    h    = bh & (H_NUM - 1);
    const size_t base = (size_t)bh * N_TOK * D_HEAD;

    // Zero score buffer: padded rows/cols must be exact 0 before the PV matmul.
    for (int i = tid; i < 64 * LSTR; i += 512) Ps[i] = 0.0f;

    // Stage Q, K (rows 0..48), row-major, stride LSTR.
    for (int i = tid; i < N_TOK * D_HEAD; i += 512) {
        int r = i >> 6, c = i & 63;
        QVs[r * LSTR + c] = Q[base + i];
        Ks [r * LSTR + c] = K[base + i];
    }
    __syncthreads();

    const int wave  = tid >> 5;          // 0..15
    const int lane  = tid & 31;
    const int r16   = lane & 15;         // row (A) / col (B,C) index within tile
    const int khalf = (lane >> 4) << 1;  // 0 or 2: K-pair held by this half-wave
    const int rt    = wave >> 2;         // 16-row tile of queries / output rows
    const int ct    = wave & 3;          // 16-col tile of keys (S) / head-dims (O)

    // ---------------- S = Q * K^T (16 x f32 WMMA, K=64) ----------------
    v8f acc = {};
    {
        const float* arow = QVs + (rt * 16 + r16) * LSTR + khalf; // Q[row][k]
        const float* brow = Ks  + (ct * 16 + r16) * LSTR + khalf; // K[col][k] == (K^T)[k][col]
        #pragma unroll
        for (int k0 = 0; k0 < D_HEAD; k0 += 4) {
            v2f a; a.x = arow[k0]; a.y = arow[k0 + 1];
            v2f b; b.x = brow[k0]; b.y = brow[k0 + 1];
            acc = __builtin_amdgcn_wmma_f32_16x16x4_f32(
                false, a, false, b, (short)0, acc, false, false);
        }
    }
    // Scale + relative-position-bias gather, write into LDS scores.
    // Clamp indices so all 16 gather loads are branchless and pipelined
    // (two waits total instead of 16 serialized round-trips); guard only
    // the final LDS stores.
    {
        const int col   = ct * 16 + r16;                 // key index
        const int rbase = rt * 16 + ((lane >> 4) << 3);  // C/D: row = j + 8*(lane>>4)
        const int colc  = (col < N_TOK) ? col : (N_TOK - 1);

        int bidx[8];
        #pragma unroll
        for (int j = 0; j < 8; ++j) {
            int rowc = rbase + j;
            rowc = (rowc < N_TOK) ? rowc : (N_TOK - 1);
            bidx[j] = rpi[rowc * N_TOK + colc];
        }
        float bias[8];
        #pragma unroll
        for (int j = 0; j < 8; ++j)
            bias[j] = rpb[bidx[j] * H_NUM + h];

        if (col < N_TOK) {
            #pragma unroll
            for (int j = 0; j < 8; ++j) {
                int row = rbase + j;
                if (row < N_TOK)
                    Ps[row * LSTR + col] = acc[j] * SCALE_Q + bias[j];
            }
        }
    }
    __syncthreads();

    // -------- Phase 2: load V^T into the dead Q buffer + row softmax --------
    // Zero the padded key columns m = 49..63 of V^T (disjoint from real loads).
    for (int i = tid; i < 64 * (64 - N_TOK); i += 512) {
        int d = i / (64 - N_TOK), m = N_TOK + i % (64 - N_TOK);
        QVs[d * LSTR + m] = 0.0f;
    }
    // V is (m, d) row-major in global; store transposed: QVs[d][m] = V[m][d].
    for (int i = tid; i < N_TOK * D_HEAD; i += 512) {
        int m = i >> 6, d = i & 63;
        QVs[d * LSTR + m] = V[base + i];
    }
    // Softmax: 8 lanes per row, shuffle reductions within 8-lane groups.
    {
        int row = tid >> 3, l8 = tid & 7;
        if (row < N_TOK) {
            float* prow = Ps + row * LSTR;
            float mx = -3.0e38f;
            for (int c = l8; c < N_TOK; c += 8) mx = fmaxf(mx, prow[c]);
            #pragma unroll
            for (int off = 4; off > 0; off >>= 1)
                mx = fmaxf(mx, __shfl_xor(mx, off, 8));
            float sum = 0.0f;
            for (int c = l8; c < N_TOK; c += 8) {
                float e = __expf(prow[c] - mx);
                prow[c] = e;
                sum += e;
            }
            #pragma unroll
            for (int off = 4; off > 0; off >>= 1)
                sum += __shfl_xor(sum, off, 8);
            float inv = 1.0f / sum;
            for (int c = l8; c < N_TOK; c += 8) prow[c] *= inv;
        }
    }
    __syncthreads();

    // ---------------- O = P * V (16 x f32 WMMA, K=64 padded) ----------------
    v8f o = {};
    {
        const float* parow = Ps  + (rt * 16 + r16) * LSTR + khalf; // P[row][m]
        const float* vtrow = QVs + (ct * 16 + r16) * LSTR + khalf; // V^T[d][m] == V[m][d]
        #pragma unroll
        for (int k0 = 0; k0 < 64; k0 += 4) {
            v2f a; a.x = parow[k0]; a.y = parow[k0 + 1];
            v2f b; b.x = vtrow[k0]; b.y = vtrow[k0 + 1];
            o = __builtin_amdgcn_wmma_f32_16x16x4_f32(
                false, a, false, b, (short)0, o, false, false);
        }
    }
    // Store: all 64 head-dim columns valid; guard padded query rows.
    {
        int col   = ct * 16 + r16;                  // head dim
        int rbase = rt * 16 + ((lane >> 4) << 3);
        #pragma unroll
        for (int j = 0; j < 8; ++j) {
            int row = rbase + j;
            if (row < N_TOK) out[base + row * D_HEAD + col] = o[j];
        }
    }
}

extern "C" void kernel_launch(void* const* d_in, const int* in_sizes, int n_in,
                              void* d_out, int out_size, void* d_ws, size_t ws_size,
                              hipStream_t stream) {
    const float* Q   = (const float*)d_in[0];
    const float* K   = (const float*)d_in[1];
    const float* V   = (const float*)d_in[2];
    const float* rpb = (const float*)d_in[3];
    const int*   rpi = (const int*)d_in[4];
    float* out = (float*)d_out;

    int BH = in_sizes[0] / (N_TOK * D_HEAD);   // 2048 * 16 = 32768
    swin_attn_f32wmma<<<BH, 512, 0, stream>>>(Q, K, V, rpb, rpi, out);
}